// MTAD_GAT_44813688766796
// MI455X (gfx1250) — compile-verified
//
#include <hip/hip_runtime.h>
#include <hip/hip_bf16.h>
#include <math.h>

// ---------------------------------------------------------------------------
// MTAD-GAT forward for MI455X (gfx1250, wave32, WMMA).
//
// Cost model: w_ih0 is 3072x68640 f32 = 843 MB -> pure HBM-bandwidth problem
// (~36us floor @ 23.3 TB/s). All GEMVs use V_WMMA_F32_16X16X4_F32:
// 16 rows per block, K split 4 ways across gridDim.y and 8 ways across waves.
// The K loop is MANUALLY unrolled x4 with 4 independent accumulators so the
// compiler issues 8 global loads back-to-back per iteration (deep MLP), plus
// a global_prefetch of the W stream. Reductions are fixed-order
// (deterministic across graph replays).
// ---------------------------------------------------------------------------

#define FEATS 32
#define NNODE 33
#define HID   1024          // FEATS*FEATS
#define IN_GRU 68640        // 33*32*65
#define NEG_SLOPE 0.2f
#define KSPLIT 4

typedef __attribute__((ext_vector_type(2))) float v2f;
typedef __attribute__((ext_vector_type(8))) float v8f;

// ---------------------------------------------------------------------------
// Kernel 1: GAT attention coefficients (closed form on the star graph).
// er[0] is always x[0]*Sb with x[0]==0, so e[n] = leaky(x[n]*Sa).
// Thread t<32: feature-GAT head t -> cF[t]. Thread t>=32: time-GAT row t-32.
// ---------------------------------------------------------------------------
__global__ void gat_coef_kernel(const float* __restrict__ data,
                                const float* __restrict__ Wf,
                                const float* __restrict__ alf,
                                const float* __restrict__ Wt,
                                const float* __restrict__ alt,
                                float* __restrict__ cF,
                                float* __restrict__ cT) {
    int t = threadIdx.x;            // 64 threads, 1 block
    if (t >= 64) return;
    bool isF = (t < 32);
    int h = isF ? t : (t - 32);

    float Sa = 0.f;
    if (isF) {
        for (int o = 0; o < FEATS; ++o) Sa += Wf[h * FEATS + o] * alf[h * FEATS + o];
    } else {
        for (int o = 0; o < FEATS; ++o) Sa += Wt[o] * alt[o];
    }

    float e[NNODE];
    float m = -1e30f;
    for (int n = 0; n < NNODE; ++n) {
        float xv = (n == 0) ? 0.f
                 : (isF ? data[(FEATS - 1) * FEATS + (n - 1)]   // d[-1][n-1]
                        : data[h * FEATS + (n - 1)]);           // d[h][n-1]
        float v = xv * Sa;
        v = (v >= 0.f) ? v : NEG_SLOPE * v;
        e[n] = v;
        m = fmaxf(m, v);
    }
    float s = 0.f, cx = 0.f;
    for (int n = 0; n < NNODE; ++n) {
        float a = __expf(e[n] - m);
        s += a;
        float xv = (n == 0) ? 0.f
                 : (isF ? data[(FEATS - 1) * FEATS + (n - 1)]
                        : data[h * FEATS + (n - 1)]);
        cx += a * xv;
    }
    cx /= s;
    if (isF) cF[h] = cx; else cT[h] = cx;
}

// ---------------------------------------------------------------------------
// Kernel 2: assemble x_in[33][32][65] = concat(d3, feat_r, time_r, axis=2).
// ---------------------------------------------------------------------------
__global__ void assemble_x_kernel(const float* __restrict__ data,
                                  const float* __restrict__ Wf,
                                  const float* __restrict__ bf,
                                  const float* __restrict__ Wt,
                                  const float* __restrict__ bt,
                                  const float* __restrict__ cF,
                                  const float* __restrict__ cT,
                                  float* __restrict__ x_in) {
    int idx = blockIdx.x * 256 + threadIdx.x;
    if (idx >= IN_GRU) return;
    int n = idx / (FEATS * 65);
    int r = idx % (FEATS * 65);
    int j = r / 65;
    int c = r % 65;
    float v;
    if (c == 0) {                                        // d3[n][j]
        v = (n == 0) ? 0.f : data[(n - 1) * FEATS + j];
    } else if (c <= FEATS) {                             // feat_r[n][j][c-1]
        int o = c - 1;
        float xv = (n == 0) ? cF[j] : data[(FEATS - 1) * FEATS + (n - 1)];
        v = xv * Wf[j * FEATS + o] + bf[j * FEATS + o];
    } else {                                             // time_r[n][j][c-33]
        int o = c - 33;
        float xv = (n == 0) ? cT[j] : data[j * FEATS + (n - 1)];
        v = xv * Wt[o] + bt[o];
    }
    x_in[idx] = v;
}

// ---------------------------------------------------------------------------
// Kernel 3: WMMA GEMV partial.  part[by][row] = sum over this block's K-slice.
// Block = 256 threads = 8 waves; 16 rows per blockIdx.x; K interleaved in
// 4-float chunks across (blockIdx.y*8 + wave) -> stride 128 floats per wave.
// A fragment (16x4 f32): lane m=lane&15 is row, half=lane>>4 selects K
// {0,1}/{2,3} -> one b64 load per lane, a contiguous per-row stream.
// B fragment: x broadcast to all 16 columns -> every lane's accumulator holds
// valid row dot-products (rows 0..7 in half 0, rows 8..15 in half 1).
// Manual x4 unroll, 4 independent accumulators -> 8 loads in flight per wave.
// ---------------------------------------------------------------------------
#define KSTRIDE (4 * 8 * KSPLIT)    // 128 floats per global-wave step

__global__ void gemv_wmma_kernel(const float* __restrict__ W,
                                 const float* __restrict__ x,
                                 float* __restrict__ part,
                                 int K, int R) {
    const int lane = threadIdx.x & 31;
    const int wave = threadIdx.x >> 5;
    const int row0 = blockIdx.x * 16;
    const int m    = lane & 15;
    const int half = lane >> 4;
    const int koff = half * 2;                 // 0 or 2
    const int gw   = blockIdx.y * 8 + wave;    // 0..31 global wave slice

    const float* __restrict__ Wp = W + (size_t)(row0 + m) * (size_t)K + koff;
    const float* __restrict__ xp = x + koff;

    v8f a0 = {0.f, 0.f, 0.f, 0.f, 0.f, 0.f, 0.f, 0.f};
    v8f a1 = a0, a2 = a0, a3 = a0;

    int k = gw * 4;
    // Main loop: 4 independent WMMA chains, 8 b64 loads issued per iteration.
    for (; k + 3 * KSTRIDE < K; k += 4 * KSTRIDE) {
        __builtin_prefetch(Wp + k + 8 * KSTRIDE, 0, 1);  // global_prefetch_b8
        v2f wa0 = *(const v2f*)(Wp + k);
        v2f wa1 = *(const v2f*)(Wp + k + KSTRIDE);
        v2f wa2 = *(const v2f*)(Wp + k + 2 * KSTRIDE);
        v2f wa3 = *(const v2f*)(Wp + k + 3 * KSTRIDE);
        v2f xb0 = *(const v2f*)(xp + k);
        v2f xb1 = *(const v2f*)(xp + k + KSTRIDE);
        v2f xb2 = *(const v2f*)(xp + k + 2 * KSTRIDE);
        v2f xb3 = *(const v2f*)(xp + k + 3 * KSTRIDE);
        a0 = __builtin_amdgcn_wmma_f32_16x16x4_f32(false, wa0, false, xb0,
                                                   (short)0, a0, false, false);
        a1 = __builtin_amdgcn_wmma_f32_16x16x4_f32(false, wa1, false, xb1,
                                                   (short)0, a1, false, false);
        a2 = __builtin_amdgcn_wmma_f32_16x16x4_f32(false, wa2, false, xb2,
                                                   (short)0, a2, false, false);
        a3 = __builtin_amdgcn_wmma_f32_16x16x4_f32(false, wa3, false, xb3,
                                                   (short)0, a3, false, false);
    }
    // Ragged tail (K/4 slots are not a multiple of 32*4).
    for (; k < K; k += KSTRIDE) {
        v2f wa = *(const v2f*)(Wp + k);
        v2f xb = *(const v2f*)(xp + k);
        a0 = __builtin_amdgcn_wmma_f32_16x16x4_f32(false, wa, false, xb,
                                                   (short)0, a0, false, false);
    }

    // Combine the 4 accumulator chains (fixed order).
    #pragma unroll
    for (int i = 0; i < 8; ++i) a0[i] = (a0[i] + a1[i]) + (a2[i] + a3[i]);

    // Deterministic fixed-order reduction across the 8 waves of this block.
    __shared__ float red[8][16];
    if (m == 0) {                              // lanes 0 and 16 of each wave
        #pragma unroll
        for (int i = 0; i < 8; ++i) red[wave][half * 8 + i] = a0[i];
    }
    __syncthreads();
    if (threadIdx.x < 16) {
        float s = 0.f;
        #pragma unroll
        for (int w = 0; w < 8; ++w) s += red[w][threadIdx.x];
        part[(size_t)blockIdx.y * (size_t)R + row0 + threadIdx.x] = s;
    }
}

// ---------------------------------------------------------------------------
// Kernel 4: combine K-split partials + bias -> y.
// ---------------------------------------------------------------------------
__global__ void combine_kernel(const float* __restrict__ part,
                               const float* __restrict__ bias,
                               float* __restrict__ y, int R) {
    int r = blockIdx.x * 256 + threadIdx.x;
    if (r >= R) return;
    float s = bias[r];
    #pragma unroll
    for (int b = 0; b < KSPLIT; ++b) s += part[(size_t)b * R + r];
    y[r] = s;
}

// ---------------------------------------------------------------------------
// Kernel 5: GRU gate nonlinearity.  i in [0,1024).
// ---------------------------------------------------------------------------
__global__ void gru_cell_kernel(const float* __restrict__ gi,
                                const float* __restrict__ gh,
                                const float* __restrict__ hprev,
                                float* __restrict__ hout,
                                float* __restrict__ mir0,
                                float* __restrict__ mir1) {
    int i = blockIdx.x * 256 + threadIdx.x;
    if (i >= HID) return;
    float r = 1.f / (1.f + __expf(-(gi[i]          + gh[i])));
    float z = 1.f / (1.f + __expf(-(gi[HID + i]    + gh[HID + i])));
    float n = tanhf(gi[2 * HID + i] + r * gh[2 * HID + i]);
    float h = (1.f - z) * n + z * hprev[i];
    hout[i] = h;
    if (mir0) mir0[i] = h;
    if (mir1) mir1[i] = h;
}

// ---------------------------------------------------------------------------
// Host launcher.
// ---------------------------------------------------------------------------
extern "C" void kernel_launch(void* const* d_in, const int* in_sizes, int n_in,
                              void* d_out, int out_size, void* d_ws, size_t ws_size,
                              hipStream_t stream) {
    (void)in_sizes; (void)n_in; (void)out_size; (void)ws_size;

    const float* data   = (const float*)d_in[0];
    const float* hidden = (const float*)d_in[1];   // [2][1][1024]
    const float* Wf     = (const float*)d_in[2];
    const float* alf    = (const float*)d_in[3];
    // d_in[4] = arf (multiplied by x[0]==0, unused)
    const float* bf     = (const float*)d_in[5];
    const float* Wt     = (const float*)d_in[6];
    const float* alt    = (const float*)d_in[7];
    // d_in[8] = art (unused for same reason)
    const float* bt     = (const float*)d_in[9];
    const float* w_ih0  = (const float*)d_in[10];
    const float* w_hh0  = (const float*)d_in[11];
    const float* b_ih0  = (const float*)d_in[12];
    const float* b_hh0  = (const float*)d_in[13];
    const float* w_ih1  = (const float*)d_in[14];
    const float* w_hh1  = (const float*)d_in[15];
    const float* b_ih1  = (const float*)d_in[16];
    const float* b_hh1  = (const float*)d_in[17];

    float* ws   = (float*)d_ws;
    float* x_in = ws;                  // 68640
    float* cF   = x_in + IN_GRU;       // 32
    float* cT   = cF + 32;             // 32
    float* gi0  = cT + 32;             // 3072
    float* gh0  = gi0 + 3 * HID;       // 3072
    float* h0   = gh0 + 3 * HID;       // 1024
    float* gi1  = h0 + HID;            // 3072
    float* gh1  = gi1 + 3 * HID;       // 3072
    float* h1   = gh1 + 3 * HID;       // 1024
    float* part = h1 + HID;            // KSPLIT*3072

    float* out = (float*)d_out;        // [h1 | h0 | h1]

    const int R = 3 * HID;             // 3072 rows for every GEMV
    dim3 gemv_grid(R / 16, KSPLIT);
    int comb_blocks = (R + 255) / 256;

    // 1) GAT coefficients + input assembly
    gat_coef_kernel<<<1, 64, 0, stream>>>(data, Wf, alf, Wt, alt, cF, cT);
    assemble_x_kernel<<<(IN_GRU + 255) / 256, 256, 0, stream>>>(
        data, Wf, bf, Wt, bt, cF, cT, x_in);

    // 2) GRU layer 0
    gemv_wmma_kernel<<<gemv_grid, 256, 0, stream>>>(w_ih0, x_in, part, IN_GRU, R);
    combine_kernel<<<comb_blocks, 256, 0, stream>>>(part, b_ih0, gi0, R);
    gemv_wmma_kernel<<<gemv_grid, 256, 0, stream>>>(w_hh0, hidden, part, HID, R);
    combine_kernel<<<comb_blocks, 256, 0, stream>>>(part, b_hh0, gh0, R);
    gru_cell_kernel<<<(HID + 255) / 256, 256, 0, stream>>>(
        gi0, gh0, hidden, h0, out + HID, nullptr);

    // 3) GRU layer 1
    gemv_wmma_kernel<<<gemv_grid, 256, 0, stream>>>(w_ih1, h0, part, HID, R);
    combine_kernel<<<comb_blocks, 256, 0, stream>>>(part, b_ih1, gi1, R);
    gemv_wmma_kernel<<<gemv_grid, 256, 0, stream>>>(w_hh1, hidden + HID, part, HID, R);
    combine_kernel<<<comb_blocks, 256, 0, stream>>>(part, b_hh1, gh1, R);
    gru_cell_kernel<<<(HID + 255) / 256, 256, 0, stream>>>(
        gi1, gh1, hidden + HID, h1, out, out + 2 * HID);
}